// MixConLoss_816043786677
// MI455X (gfx1250) — compile-verified
//
#include <hip/hip_runtime.h>
#include <math.h>

// MixConLoss for MI455X (gfx1250), fp32 WMMA path.
//
// Problem: P=2048 pos, N0=2048 neg0, N1=2048 neg1, K=128. M=6144.
// Only rows [0,4096) of sim are reduced. Whole feature matrix = 3MB -> L2
// resident; compute bound on V_WMMA_F32_16X16X4_F32 + ~25M v_exp_f32.
// fp32 WMMA chosen over f16/bf16: 1/T=5 scaling + exp amplifies matmul error.

typedef __attribute__((ext_vector_type(2))) float v2f;
typedef __attribute__((ext_vector_type(8))) float v8f;

#define PPOS   2048
#define NN0    2048
#define NN1    2048
#define KDIM   128
#define MTOT   6144
#define ROWS   4096            // rows contributing to the loss
#define INV_T  5.0f            // 1 / 0.2
#define WAVES_PER_BLOCK 8

// ---------------------------------------------------------------------------
// Prep 1: stable partition index for neg_feat1 (obj_mask=1 first, stable),
// and zero the global accumulators. Trivial serial scan (2048 elems).
// ---------------------------------------------------------------------------
__global__ void prep_order(const int* __restrict__ objmask,
                           int* __restrict__ dstidx,
                           float* __restrict__ accum) {
    if (blockIdx.x == 0 && threadIdx.x == 0) {
        accum[0] = 0.0f;   // sum(loss*keep)
        accum[1] = 0.0f;   // sum(keep)
        int ones = 0;
        for (int i = 0; i < NN1; ++i) ones += (objmask[i] != 0);
        int o = 0, z = 0;
        for (int i = 0; i < NN1; ++i) {
            if (objmask[i] != 0) dstidx[i] = o++;
            else                 dstidx[i] = ones + z++;
        }
    }
}

// ---------------------------------------------------------------------------
// Prep 2: build all_feat = concat(pos_feat, neg_feat0, neg_feat1[order]).
// One thread per float4 (row-major, K=128 -> 32 float4 per row).
// ---------------------------------------------------------------------------
__global__ __launch_bounds__(256)
void prep_build(const float* __restrict__ pos, const float* __restrict__ n0,
                const float* __restrict__ n1, const int* __restrict__ dstidx,
                float* __restrict__ allf) {
    int idx = blockIdx.x * blockDim.x + threadIdx.x;
    if (idx >= MTOT * (KDIM / 4)) return;
    int row = idx >> 5;         // / 32
    int q   = idx & 31;
    float4 v;
    int drow = row;
    if (row < PPOS) {
        v = ((const float4*)pos)[row * 32 + q];
    } else if (row < PPOS + NN0) {
        v = ((const float4*)n0)[(row - PPOS) * 32 + q];
    } else {
        int s = row - (PPOS + NN0);
        v = ((const float4*)n1)[s * 32 + q];
        drow = PPOS + NN0 + dstidx[s];      // scatter to stable-sorted slot
    }
    ((float4*)allf)[drow * 32 + q] = v;
}

// ---------------------------------------------------------------------------
// Prep 3: per-position match key. Pos labels (0..79) and group ids (offset
// +1000, tiled over both neg halves by POSITION) never collide, so
// label_mask[i][j] == (key[i]==key[j]) and row_sum == count of key matches.
// ---------------------------------------------------------------------------
__global__ void prep_key(const int* __restrict__ labels,
                         const int* __restrict__ groups,
                         int* __restrict__ key) {
    int j = blockIdx.x * blockDim.x + threadIdx.x;
    if (j >= MTOT) return;
    int k;
    if (j < PPOS)            k = labels[j];
    else if (j < PPOS + NN0) k = 1000 + groups[j - PPOS];
    else                     k = 1000 + groups[j - PPOS - NN0];
    key[j] = k;
}

// ---------------------------------------------------------------------------
// Main: each wave owns a 16-row tile, loops over 384 column tiles.
// Per tile: 32x V_WMMA_F32_16X16X4_F32 (K=128), then per-lane flash-softmax
// update with a SINGLE exp per element: of exp(s-mn) and exp(m-mn) exactly
// one is != 1, so compute t = exp(-|s-m|) once and select.
// Final 16-lane butterfly merges per-lane partials; lanes 0/16 finish rows.
// ---------------------------------------------------------------------------
__global__ __launch_bounds__(256)
void simloss_main(const float* __restrict__ F, const int* __restrict__ key,
                  const float* __restrict__ ious, float* __restrict__ accum) {
    const int lane = threadIdx.x & 31;
    const int wave = threadIdx.x >> 5;
    const int rowbase = (blockIdx.x * WAVES_PER_BLOCK + wave) * 16;
    const int half = lane >> 4;       // 0: M/K 0..., 1: +8 rows / +2 K
    const int l15  = lane & 15;

    // A fragment (16 rows x K=128) held in registers for the whole sweep.
    // 16x4 f32 A layout: lanes 0-15 -> (K=k0,k0+1), lanes 16-31 -> (k0+2,k0+3).
    v2f a[32];
    const float* arow = F + (size_t)(rowbase + l15) * KDIM + half * 2;
#pragma unroll
    for (int t = 0; t < 32; ++t)
        a[t] = *(const v2f*)(arow + t * 4);

    int rkey[8];
#pragma unroll
    for (int r = 0; r < 8; ++r)
        rkey[r] = key[rowbase + r + 8 * half];

    float mrun[8], zrun[8], ssum[8], cnt[8];
#pragma unroll
    for (int r = 0; r < 8; ++r) {
        mrun[r] = -3.4e38f; zrun[r] = 0.0f; ssum[r] = 0.0f; cnt[r] = 0.0f;
    }

    for (int ct = 0; ct < MTOT / 16; ++ct) {
        const int colbase = ct * 16;
        const int mycol = colbase + l15;
        const float* bcol = F + (size_t)mycol * KDIM + half * 2;
        if (ct + 1 < MTOT / 16)   // WGP-scope prefetch of next B column tile
            __builtin_prefetch(F + (size_t)(mycol + 16) * KDIM, 0, 3);

        v8f c = {};
#pragma unroll
        for (int t = 0; t < 32; ++t) {
            v2f b = *(const v2f*)(bcol + t * 4);
            // D = A x B + C, fp32, 16x16x4
            c = __builtin_amdgcn_wmma_f32_16x16x4_f32(
                    false, a[t], false, b, (short)0, c, false, false);
        }

        const int ckey = key[mycol];
#pragma unroll
        for (int r = 0; r < 8; ++r) {
            const int row = rowbase + r + 8 * half;      // C/D layout
            const float s = c[r] * INV_T;
            const bool diag  = (row == mycol);
            const bool match = (ckey == rkey[r]);
            cnt[r]  += match ? 1.0f : 0.0f;              // includes diagonal
            ssum[r] += (match && !diag) ? s : 0.0f;      // excludes diagonal
            // flash-softmax update, single exp:
            //   mn = max(m, s); exp(s-mn) and exp(m-mn): one equals 1,
            //   the other equals t = exp(-|s-m|).
            const float d = s - mrun[r];
            const float t = __expf(-fabsf(d));
            const bool  gt = (d > 0.0f);                 // s is the new max
            float e     = gt ? 1.0f : t;                 // exp(s - mn)
            float scale = gt ? t : 1.0f;                 // exp(m - mn)
            e = diag ? 0.0f : e;                         // Z excludes diagonal
            zrun[r] = zrun[r] * scale + e;
            mrun[r] = fmaxf(mrun[r], s);                 // max includes diagonal
        }
    }

    // Butterfly merge across the 16 lanes of each half.
#pragma unroll
    for (int r = 0; r < 8; ++r) {
#pragma unroll
        for (int x = 1; x < 16; x <<= 1) {
            float om = __shfl_xor(mrun[r], x, 32);
            float oz = __shfl_xor(zrun[r], x, 32);
            float os = __shfl_xor(ssum[r], x, 32);
            float oc = __shfl_xor(cnt[r],  x, 32);
            float mn = fmaxf(mrun[r], om);
            zrun[r]  = zrun[r] * __expf(mrun[r] - mn) + oz * __expf(om - mn);
            mrun[r]  = mn;
            ssum[r] += os;
            cnt[r]  += oc;
        }
    }

    if (l15 == 0) {   // lanes 0 and 16 each finalize 8 rows
        float lsum = 0.0f, ksum = 0.0f;
#pragma unroll
        for (int r = 0; r < 8; ++r) {
            int row = rowbase + r + 8 * half;
            float kc = (row < PPOS) ? ((ious[row] >= 0.5f) ? 1.0f : 0.0f)
                                    : 1.0f;
            float logZ = __logf(zrun[r]);
            // plp = sum_{mask,j!=i}(s - m - logZ) / cnt
            float plp = (ssum[r] - (cnt[r] - 1.0f) * (mrun[r] + logZ)) / cnt[r];
            lsum += -plp * kc;
            ksum += kc;
        }
        atomicAdd(&accum[0], lsum);
        atomicAdd(&accum[1], ksum);
    }
}

__global__ void finalize(const float* __restrict__ accum,
                         float* __restrict__ out) {
    if (blockIdx.x == 0 && threadIdx.x == 0)
        out[0] = accum[0] / accum[1];
}

// ---------------------------------------------------------------------------
// Launcher. Inputs (setup_inputs order):
//  0 pos_feat (2048x128 f32)  1 pos_labels (2048 int)  2 ious_pos (2048 f32)
//  3 neg_feat0 (2048x128 f32) 4 neg_feat1 (2048x128 f32)
//  5 neg_group_ids (2048 int) 6 neg_obj_mask (2048 int)
// d_out: 1 float.
// ---------------------------------------------------------------------------
extern "C" void kernel_launch(void* const* d_in, const int* in_sizes, int n_in,
                              void* d_out, int out_size, void* d_ws, size_t ws_size,
                              hipStream_t stream) {
    const float* pos_feat = (const float*)d_in[0];
    const int*   labels   = (const int*)  d_in[1];
    const float* ious     = (const float*)d_in[2];
    const float* neg0     = (const float*)d_in[3];
    const float* neg1     = (const float*)d_in[4];
    const int*   groups   = (const int*)  d_in[5];
    const int*   objmask  = (const int*)  d_in[6];

    char* ws = (char*)d_ws;
    float* accum  = (float*)ws;              // 2 floats
    int*   dstidx = (int*)(ws + 1024);       // 2048 ints
    int*   key    = (int*)(ws + 16384);      // 6144 ints
    float* allf   = (float*)(ws + 65536);    // 6144*128 floats (3 MB)

    prep_order<<<1, 32, 0, stream>>>(objmask, dstidx, accum);

    int nv4 = MTOT * (KDIM / 4);
    prep_build<<<(nv4 + 255) / 256, 256, 0, stream>>>(pos_feat, neg0, neg1,
                                                      dstidx, allf);
    prep_key<<<(MTOT + 255) / 256, 256, 0, stream>>>(labels, groups, key);

    // 4096 rows / (16 rows per wave * 8 waves per block) = 32 blocks
    simloss_main<<<ROWS / (16 * WAVES_PER_BLOCK), 32 * WAVES_PER_BLOCK, 0,
                   stream>>>(allf, key, ious, accum);

    finalize<<<1, 32, 0, stream>>>(accum, (float*)d_out);
}